// CorrelationLayer_14259291423008
// MI455X (gfx1250) — compile-verified
//
#include <hip/hip_runtime.h>

typedef __attribute__((ext_vector_type(2))) float v2f;
typedef __attribute__((ext_vector_type(4))) float v4f;
typedef __attribute__((ext_vector_type(8))) float v8f;

#define MAXD 10
#define KS   21            // 2*MAXD+1
#define ND   (KS*KS)       // 441
#define Bn   4
#define Cn   64
#define Hn   128
#define Wn   128
#define HW   (Hn*Wn)
#define CH_STRIDE (Hn*Wn)  // channel stride in elements
// feat2 LDS: 32 channel-pairs, interleaved columns (16 pad | 128 | 16 pad)
#define LDSW 160
#define P2   (2*LDSW + 32)   // 352 floats; P2 mod 64 == 32 -> hi half-wave banks disjoint
// output staging: [21 j-rows][132] ; 132 mod 64 == 4 -> scatter writes conflict-free,
// rows 16B-aligned for b128 readout.
#define PW   132

__launch_bounds__(256)
__global__ void corr_wmma_kernel(const float* __restrict__ feat1,
                                 const float* __restrict__ feat2,
                                 float* __restrict__ out)
{
    __shared__ float s2[32 * P2];       // ~45 KB feat2 row (pair-interleaved, padded)
    __shared__ float s_out[KS * PW];    // ~10.8 KB band staging [j][w]

    const int tid  = threadIdx.x;
    const int wave = tid >> 5;          // 8 waves, one 16-pixel M-tile each
    const int lane = tid & 31;
    const int hi   = lane >> 4;         // half-wave select
    const int lo   = lane & 15;
    const int h    = blockIdx.x;
    const int b    = blockIdx.y;
    const int w0   = wave * 16;

    // ---- Preload A fragments once: feat1[b, :, h, w0+lo], 16 K-steps of K=4.
    // 16x4 f32 A layout: lanes 0-15 M=lane (VGPR0=K0, VGPR1=K1); lanes 16-31 (K2,K3).
    v2f a[16];
    {
        const float* f1 = feat1 + ((b * Cn) * Hn + h) * Wn + w0 + lo;
        #pragma unroll
        for (int kk = 0; kk < 16; ++kk) {
            const int c0 = 4 * kk + 2 * hi;
            a[kk].x = f1[(c0    ) * CH_STRIDE];
            a[kk].y = f1[(c0 + 1) * CH_STRIDE];
        }
    }

    // ---- One-time zero of the pad columns (cols 0..15 and 144..159 of each pair-row).
    #pragma unroll
    for (int it = 0; it < 4; ++it) {
        const int idx = it * 256 + tid;       // 1024 pad slots
        const int p = idx >> 5;               // pair 0..31
        const int k = idx & 31;               // 0..31
        const int col = (k < 16) ? k : (128 + k);   // 0..15 or 144..159
        *(v2f*)(s2 + p * P2 + 2 * col) = (v2f){0.f, 0.f};
    }

    const v8f vzero = {0.f,0.f,0.f,0.f,0.f,0.f,0.f,0.f};
    const float scale = 1.0f / (float)Cn;

    for (int i = 0; i < KS; ++i) {
        const int h2 = h + i - MAXD;
        const bool rowOK = (unsigned)h2 < (unsigned)Hn;   // block-uniform

        __syncthreads();  // prior iteration's s2 reads complete before refill
        if (rowOK) {
            // Fill interior: channel-pair interleaved, ds_store_b64, coalesced in w.
            const float* f2 = feat2 + (b * Cn * Hn + h2) * Wn;  // c=0 base, +c*CH_STRIDE
            #pragma unroll
            for (int it = 0; it < 16; ++it) {        // 32 pairs * 128 w / 256 thr
                const int idx = it * 256 + tid;
                const int p = idx >> 7;              // pair 0..31
                const int w = idx & 127;
                v2f d;
                d.x = f2[(2 * p    ) * CH_STRIDE + w];
                d.y = f2[(2 * p + 1) * CH_STRIDE + w];
                *(v2f*)(s2 + p * P2 + 2 * (w + 16)) = d;
            }
        }
        // Prefetch next feat2 row toward caches (256 threads x 128B lines covers it)
        {
            const int h2n = h2 + 1;
            if ((i + 1 < KS) && ((unsigned)h2n < (unsigned)Hn)) {
                const int c = tid >> 2, seg = tid & 3;
                __builtin_prefetch(feat2 + (b * Cn * Hn + h2n) * Wn + c * CH_STRIDE + seg * 32, 0, 0);
            }
        }
        __syncthreads();

        v8f acc[3];
        #pragma unroll
        for (int t = 0; t < 3; ++t) acc[t] = vzero;

        if (rowOK) {
            // B fragment for k-step kk, tile t: channels {4kk+2hi, 4kk+2hi+1} = pair (2kk+hi),
            // column (w0+lo) + t*16 in padded coords -> single ds_load_b64, bank-conflict-free.
            const int colOff = 2 * (w0 + lo);
            #pragma unroll
            for (int kk = 0; kk < 16; ++kk) {
                const float* sp = s2 + (2 * kk + hi) * P2 + colOff;
                const v2f b0 = *(const v2f*)(sp);        // tile t=0 (base w0-16)
                const v2f b1 = *(const v2f*)(sp + 32);   // tile t=1 (base w0)
                const v2f b2 = *(const v2f*)(sp + 64);   // tile t=2 (base w0+16)
                acc[0] = __builtin_amdgcn_wmma_f32_16x16x4_f32(
                    false, a[kk], false, b0, (short)0, acc[0], false, false);
                acc[1] = __builtin_amdgcn_wmma_f32_16x16x4_f32(
                    false, a[kk], false, b1, (short)0, acc[1], false, false);
                acc[2] = __builtin_amdgcn_wmma_f32_16x16x4_f32(
                    false, a[kk], false, b2, (short)0, acc[2], false, false);
            }
        }

        // ---- Band extraction into LDS staging: D[M,N], M=r+8*hi, N=lo;
        //      j = N - M + 10 + 16*(t-1); s_out[j][w0+M] = D * scale.
        #pragma unroll
        for (int t = 0; t < 3; ++t) {
            union { v8f v; float f[8]; } u;
            u.v = acc[t];
            #pragma unroll
            for (int r = 0; r < 8; ++r) {
                const int M = r + 8 * hi;
                const int j = lo - M + MAXD + 16 * (t - 1);
                if ((unsigned)j < (unsigned)KS) {
                    s_out[j * PW + w0 + M] = u.f[r] * scale;
                }
            }
        }
        __syncthreads();

        // ---- Coalesced writeback: one wave per j-row, 32 lanes x b128 = 512B/instr.
        {
            float* outbase = out + (b * ND + i * KS) * HW + h * Wn;
            #pragma unroll
            for (int it = 0; it < 3; ++it) {
                const int idx = it * 256 + tid;       // 21 rows * 32 quads = 672
                if (idx < KS * 32) {
                    const int j = idx >> 5;
                    const int q = (idx & 31) * 4;
                    const v4f d = *(const v4f*)(s_out + j * PW + q);
                    *(v4f*)(outbase + j * HW + q) = d;
                }
            }
        }
    }
}

extern "C" void kernel_launch(void* const* d_in, const int* in_sizes, int n_in,
                              void* d_out, int out_size, void* d_ws, size_t ws_size,
                              hipStream_t stream) {
    (void)in_sizes; (void)n_in; (void)d_ws; (void)ws_size; (void)out_size;
    const float* feat1 = (const float*)d_in[0];
    const float* feat2 = (const float*)d_in[1];
    float* out = (float*)d_out;

    dim3 grid(Hn, Bn);   // one block per (h, b): 8 waves cover the full W=128 row
    dim3 block(256);
    corr_wmma_kernel<<<grid, block, 0, stream>>>(feat1, feat2, out);
}